// ScriptKalmanFilter_54374285967482
// MI455X (gfx1250) — compile-verified
//
#include <hip/hip_runtime.h>
#include <hip/hip_bf16.h>

#define G 2048
#define T 200
#define S 8
#define M 2

typedef float v8f __attribute__((ext_vector_type(8)));
typedef float v2f __attribute__((ext_vector_type(2)));

// D += A(16x16) * B(16x16)   (or A * B^T when TRB), via 4 chained V_WMMA_F32_16X16X4_F32.
// A,B live in LDS (row-major 16x16).  acc is in C/D layout:
//   vgpr r, lane l  ->  element [r + 8*(l>>4)][l & 15]
// A operand layout (ISA 7.12.2, 32-bit A 16x4): lane l, vgpr v -> A[l&15][2*(l>>4)+v]
template <bool TRB>
__device__ __forceinline__ v8f wmma16x16(const float* A, const float* B, v8f acc,
                                         unsigned lane) {
  const unsigned m  = lane & 15u;
  const unsigned hk = (lane >> 4) << 1;  // 0 or 2
#pragma unroll
  for (unsigned k0 = 0; k0 < 16; k0 += 4) {
    const unsigned k = k0 + hk;
    v2f a, b;
    a.x = A[m * 16 + k];
    a.y = A[m * 16 + k + 1];
    if (!TRB) {
      b.x = B[k * 16 + m];
      b.y = B[(k + 1) * 16 + m];
    } else {  // B operand = B^T : Bop[k][n] = B[n][k]
      b.x = B[m * 16 + k];
      b.y = B[m * 16 + k + 1];
    }
    acc = __builtin_amdgcn_wmma_f32_16x16x4_f32(false, a, false, b, (short)0, acc,
                                                false, false);
  }
  return acc;
}

__device__ __forceinline__ void wmma_store(v8f acc, float* D, unsigned lane) {
  const unsigned n  = lane & 15u;
  const unsigned rh = (lane >> 4) << 3;  // 0 or 8
#pragma unroll
  for (unsigned r = 0; r < 8; ++r) D[(r + rh) * 16 + n] = acc[r];
}

// Per-lane async gather: 16B from global -> 16B into LDS at this lane's LDS offset.
// Tracked by ASYNCcnt; fence with s_wait_asynccnt before consuming.
__device__ __forceinline__ void async_gather_b128(const float* __restrict__ gsrc,
                                                  const float* lds_dst) {
  const unsigned loff = (unsigned)(uintptr_t)lds_dst;  // flat->LDS: low 32 bits
  asm volatile("global_load_async_to_lds_b128 %0, %1, off"
               :
               : "v"(loff), "v"(gsrc)
               : "memory");
}

// One wave (32 lanes) runs the T-step Kalman recursion for TWO groups packed
// block-diagonally into 16x16 tiles.  blockDim = 32 (single-wave workgroup,
// so barriers degenerate to NOPs and all LDS hazards are in-wave waitcnts).
__global__ __launch_bounds__(32) void kf_scan(
    const float* __restrict__ obs,    // [G,T,M]
    const float* __restrict__ mean0,  // [G,S]
    const float* __restrict__ cov0,   // [G,S,S]
    const float* __restrict__ Fs,     // [T,G,S,S]
    const float* __restrict__ Hs,     // [T,G,M,S]
    const float* __restrict__ Qs,     // [T,G,S,S]
    const float* __restrict__ Rs,     // [T,G,M,M]
    float* __restrict__ outMean,      // [G,T,S]
    float* __restrict__ outCov)       // [G,T,S,S]
{
  __shared__ __align__(16) float Fbd[256];  // block-diag F
  __shared__ __align__(16) float Qbd[256];  // block-diag Q
  __shared__ __align__(16) float COV[256];  // block-diag P
  __shared__ __align__(16) float T1s[256];  // scratch (A1 = I-KH, then F*P_u)
  __shared__ __align__(16) float T2s[256];  // scratch (P_u)
  __shared__ float Hsh[2][16];
  __shared__ float Rsh[2][4];
  __shared__ float ysh[2][2];
  __shared__ float meanv[16];
  __shared__ float resid[2][2];
  __shared__ float PHt[2][16];
  __shared__ float Ssh[2][4];
  __shared__ float Sinv[2][4];
  __shared__ float Ksh[2][16];

  const unsigned lane = threadIdx.x;  // 0..31
  const unsigned grp  = lane >> 4;    // which of the 2 groups
  const unsigned li   = lane & 15u;
  const unsigned g    = blockIdx.x * 2 + grp;

  // zero the 16x16 tiles once; only diagonal 8x8 blocks get rewritten later,
  // so the off-diagonal zeros are a loop invariant.
  for (unsigned i = lane; i < 256; i += 32) {
    Fbd[i] = 0.f; Qbd[i] = 0.f; COV[i] = 0.f; T1s[i] = 0.f; T2s[i] = 0.f;
  }

  // this lane's slot inside the diagonal 8x8 block (float4 granularity)
  const unsigned idx4 = li * 4;                 // 0..60
  const unsigned bi   = idx4 >> 3;              // row in 8x8 block
  const unsigned bj   = idx4 & 7u;              // col in 8x8 block (0 or 4)
  const unsigned bd   = (bi + 8 * grp) * 16 + (bj + 8 * grp);  // block-diag offset

  // seed P <- cov0
  {
    const float4 v = *reinterpret_cast<const float4*>(&cov0[(size_t)g * 64 + idx4]);
    *reinterpret_cast<float4*>(&COV[bd]) = v;
  }
  if (lane < 16) {
    const unsigned gg = blockIdx.x * 2 + (lane >> 3);
    meanv[lane] = mean0[(size_t)gg * 8 + (lane & 7u)];
  }

  for (int t = 0; t < T; ++t) {
    const size_t tg = (size_t)t * G + g;

    // ---- kick off async gathers of F,Q into the block-diag LDS tiles.
    // DS and ASYNC counters are unordered w.r.t. each other, so make sure the
    // previous step's DS reads of Fbd/Qbd have retired before overwriting.
    asm volatile("s_wait_dscnt 0x0" ::: "memory");
    async_gather_b128(&Fs[tg * 64 + idx4], &Fbd[bd]);
    async_gather_b128(&Qs[tg * 64 + idx4], &Qbd[bd]);

    // ---- emit prior (mean, cov) : the carry BEFORE assimilation ----
    if (lane < 16) {
      const unsigned gg = blockIdx.x * 2 + (lane >> 3);
      outMean[((size_t)gg * T + t) * 8 + (lane & 7u)] = meanv[lane];
    }
    {
      const float4 v = *reinterpret_cast<const float4*>(&COV[bd]);
      *reinterpret_cast<float4*>(&outCov[((size_t)g * T + t) * 64 + idx4]) = v;
    }

    // ---- small step inputs ----
    Hsh[grp][li] = Hs[tg * 16 + li];
    if (li < 4) Rsh[grp][li] = Rs[tg * 4 + li];
    if (li < 2) ysh[grp][li] = obs[((size_t)g * T + t) * 2 + li];
    if (t + 1 < T) {  // warm L2 for next step's big operands (global_prefetch_b8)
      __builtin_prefetch(&Fs[((size_t)(t + 1) * G + g) * 64 + idx4], 0, 1);
      __builtin_prefetch(&Qs[((size_t)(t + 1) * G + g) * 64 + idx4], 0, 1);
    }

    // ---- resid = y - H*mean ----
    if (li < 2) {
      float d = 0.f;
#pragma unroll
      for (int s = 0; s < 8; ++s) d += Hsh[grp][li * 8 + s] * meanv[grp * 8 + s];
      resid[grp][li] = ysh[grp][li] - d;
    }
    // ---- PHt = P * H^T  [8x2], one (s,m) per lane ----
    {
      const unsigned s = li >> 1, m = li & 1u;
      float d = 0.f;
#pragma unroll
      for (int j = 0; j < 8; ++j)
        d += COV[(s + 8 * grp) * 16 + j + 8 * grp] * Hsh[grp][m * 8 + j];
      PHt[grp][s * 2 + m] = d;
    }
    // ---- Ssys = H*PHt + R  [2x2] ----
    if (li < 4) {
      const unsigned mi = li >> 1, ni = li & 1u;
      float d = Rsh[grp][mi * 2 + ni];
#pragma unroll
      for (int s = 0; s < 8; ++s) d += Hsh[grp][mi * 8 + s] * PHt[grp][s * 2 + ni];
      Ssh[grp][mi * 2 + ni] = d;
    }
    // ---- analytic 2x2 inverse ----
    if (li < 4) {
      const float a = Ssh[grp][0], b = Ssh[grp][1], c = Ssh[grp][2], dd = Ssh[grp][3];
      const float inv = 1.0f / (a * dd - b * c);
      const float e = (li == 0) ? dd : (li == 1) ? -b : (li == 2) ? -c : a;
      Sinv[grp][li] = e * inv;
    }
    // ---- K = PHt * Sinv  [8x2] ----
    {
      const unsigned s = li >> 1, m = li & 1u;
      Ksh[grp][s * 2 + m] = PHt[grp][s * 2 + 0] * Sinv[grp][0 * 2 + m] +
                            PHt[grp][s * 2 + 1] * Sinv[grp][1 * 2 + m];
    }
    // ---- mean_u = mean + K*resid ----
    float mu = 0.f;
    if (li < 8)
      mu = meanv[grp * 8 + li] + Ksh[grp][li * 2 + 0] * resid[grp][0] +
           Ksh[grp][li * 2 + 1] * resid[grp][1];
    if (li < 8) meanv[grp * 8 + li] = mu;

    // ---- A1 = I - K*H into T1 diag blocks ----
    {
      float4 v;
      float* pv = reinterpret_cast<float*>(&v);
#pragma unroll
      for (unsigned jj = 0; jj < 4; ++jj) {
        const unsigned j = bj + jj;
        float d = (bi == j) ? 1.f : 0.f;
        d -= Ksh[grp][bi * 2 + 0] * Hsh[grp][0 * 8 + j] +
             Ksh[grp][bi * 2 + 1] * Hsh[grp][1 * 8 + j];
        pv[jj] = d;
      }
      *reinterpret_cast<float4*>(&T1s[bd]) = v;
    }

    // ---- F,Q tiles must be resident before the WMMA chain ----
    asm volatile("s_wait_asynccnt 0x0" ::: "memory");

    // ---- P_u = A1 * P  (4x v_wmma_f32_16x16x4_f32, block-diagonal) ----
    v8f acc = {};
    acc = wmma16x16<false>(T1s, COV, acc, lane);
    wmma_store(acc, T2s, lane);

    // ---- t1 = F * P_u ----
    v8f acc2 = {};
    acc2 = wmma16x16<false>(Fbd, T2s, acc2, lane);
    wmma_store(acc2, T1s, lane);

    // ---- P_p = t1 * F^T + Q   (C accumulator = Q read from LDS in D-layout;
    //      off-diagonal zeros come from the zero-filled LDS tile) ----
    v8f acc3;
    {
      const unsigned rh = grp << 3;
#pragma unroll
      for (unsigned r = 0; r < 8; ++r) acc3[r] = Qbd[(r + rh) * 16 + li];
    }
    acc3 = wmma16x16<true>(T1s, Fbd, acc3, lane);
    wmma_store(acc3, COV, lane);

    // ---- mean_p = F * mean_u ----
    float mp = 0.f;
    if (li < 8) {
#pragma unroll
      for (int j = 0; j < 8; ++j)
        mp += Fbd[(li + 8 * grp) * 16 + j + 8 * grp] * meanv[grp * 8 + j];
    }
    if (li < 8) meanv[grp * 8 + li] = mp;
  }
}

// Rs : [T,G,2,2] -> [G,T,2,2]  via LDS tile transpose (both sides coalesced).
#define TT ((T + 15) / 16)  // 13 tiles along T
__global__ __launch_bounds__(256) void transposeR(const float4* __restrict__ src,
                                                  float4* __restrict__ dst) {
  __shared__ float4 tile[16][17];  // +1 pad -> conflict-free b128 banks
  const unsigned tx = threadIdx.x & 15u, ty = threadIdx.x >> 4;
  const unsigned tTile = blockIdx.x % TT;
  const unsigned gTile = blockIdx.x / TT;
  const unsigned t0 = tTile * 16, g0 = gTile * 16;

  const unsigned ts = t0 + ty, gs = g0 + tx;
  if (ts < T) tile[ty][tx] = src[(size_t)ts * G + gs];
  __syncthreads();
  const unsigned td = t0 + tx, gd = g0 + ty;
  if (td < T) dst[(size_t)gd * T + td] = tile[tx][ty];
}

// Hs : [T,G,2,8] -> [G,T,2,8]   (64 bytes per (t,g) on both sides)
__global__ void transposeH(const float4* __restrict__ src, float4* __restrict__ dst) {
  const unsigned idx = blockIdx.x * blockDim.x + threadIdx.x;
  if (idx >= T * G) return;
  const unsigned t = idx / G, g = idx % G;
  const float4* s = src + (size_t)idx * 4;
  float4* d = dst + ((size_t)g * T + t) * 4;
  d[0] = s[0]; d[1] = s[1]; d[2] = s[2]; d[3] = s[3];
}

extern "C" void kernel_launch(void* const* d_in, const int* in_sizes, int n_in,
                              void* d_out, int out_size, void* d_ws, size_t ws_size,
                              hipStream_t stream) {
  (void)in_sizes; (void)n_in; (void)out_size; (void)d_ws; (void)ws_size;
  const float* obs   = (const float*)d_in[0];
  const float* mean0 = (const float*)d_in[1];
  const float* cov0  = (const float*)d_in[2];
  const float* Fs    = (const float*)d_in[3];
  const float* Hs    = (const float*)d_in[4];
  const float* Qs    = (const float*)d_in[5];
  const float* Rs    = (const float*)d_in[6];

  float* out     = (float*)d_out;
  float* outMean = out;                                   // [G,T,S]
  float* outCov  = outMean + (size_t)G * T * S;           // [G,T,S,S]
  float* outR    = outCov + (size_t)G * T * S * S;        // [G,T,M,M]
  float* outH    = outR + (size_t)G * T * M * M;          // [G,T,M,S]

  kf_scan<<<G / 2, 32, 0, stream>>>(obs, mean0, cov0, Fs, Hs, Qs, Rs, outMean, outCov);
  transposeR<<<TT * (G / 16), 256, 0, stream>>>((const float4*)Rs, (float4*)outR);
  transposeH<<<(T * G + 255) / 256, 256, 0, stream>>>((const float4*)Hs, (float4*)outH);
}